// GResNet_83837761618527
// MI455X (gfx1250) — compile-verified
//
#include <hip/hip_runtime.h>
#include <hip/hip_bf16.h>

// ---------------------------------------------------------------------------
// GCN-ResNet for MI455X (gfx1250, wave32, WMMA).
//   out = act( x @ Ws + (sum of 16 gathered neighbour rows) @ Wn + bias )
// One kernel per layer; gather fused into the A-operand staging of the GEMM.
// FP32 path via V_WMMA_F32_16X16X4_F32 (keeps reference numerics).
// B staged k-pair-interleaved in LDS so each B fragment is one ds_load_b64;
// per k-step the 8 B fragments are loaded as a batch before the WMMA burst.
// ---------------------------------------------------------------------------

typedef __attribute__((ext_vector_type(2))) float v2f;
typedef __attribute__((ext_vector_type(8))) float v8f;

#define DHID   256      // hidden width (cols of every GEMM)
#define MB     64       // nodes per workgroup
#define KC     32       // K-chunk staged per iteration
#define APAD   34       // LDS row stride for A chunk (even -> aligned b64 pairs)
#define BPAIR  544      // words per interleaved kp row: 512 data + 32 pad
                        // (BPAIR % 64 == 32 -> half-wave bank sets disjoint)

__global__ __launch_bounds__(256)
void gcn_layer_kernel(const float* __restrict__ xin,   // [N, K1]
                      const int*   __restrict__ nb,    // [N, 16]
                      const float* __restrict__ Ws,    // [K1, 256]
                      const float* __restrict__ Wn,    // [K1, 256]
                      const float* __restrict__ bias,  // [256]
                      const float* __restrict__ res,   // [N, 256] or null
                      float*       __restrict__ out,   // [N, 256]
                      int N, int K1)
{
    __shared__ float Alds[MB * APAD];            //  8704 B
    __shared__ float Blds[(KC / 2) * BPAIR];     // 34816 B  (pair-interleaved)
    __shared__ int   NbLds[MB * 16];             //  4096 B  (pre-scaled offsets)

    const int tid   = threadIdx.x;
    const int lane  = tid & 31;
    const int wave  = tid >> 5;              // 0..7
    const int rt    = wave & 3;              // row tile within block (16 rows)
    const int ch    = wave >> 2;             // column half (128 cols)
    const int node0 = blockIdx.x * MB;

    // ---- preload neighbour indices, pre-multiplied by row stride ------------
    for (int idx = tid; idx < MB * 16; idx += 256) {
        const int node = node0 + (idx >> 4);
        NbLds[idx] = (node < N) ? nb[(size_t)node * 16 + (idx & 15)] * K1 : 0;
    }

    // ---- accumulators: 8 col-tiles of 16x16 per wave ------------------------
    v8f acc[8];
#pragma unroll
    for (int t = 0; t < 8; ++t)
        acc[t] = v8f{0.f, 0.f, 0.f, 0.f, 0.f, 0.f, 0.f, 0.f};

    const bool vec2    = (K1 & 1) == 0;      // hidden layers: float2 gather
    const int  nchunks = (K1 + KC - 1) / KC;

    for (int half = 0; half < 2; ++half) {
        const float* __restrict__ W = half ? Wn : Ws;

        for (int c = 0; c < nchunks; ++c) {
            const int k0 = c * KC;
            __syncthreads();

            // ---- stage A chunk [64 x 32]: self rows or fused 16-way gather --
            if (vec2) {
                // float2 per lane: lanes 0..15 cover k pairs of row m,
                // lanes 16..31 cover row m+1; 4 passes cover 64 rows.
                const int kh = lane & 15;
                const int mr = lane >> 4;
                const int kk = k0 + 2 * kh;
#pragma unroll
                for (int pass = 0; pass < 4; ++pass) {
                    const int m    = pass * 16 + wave * 2 + mr;
                    const int node = node0 + m;
                    float2 v = make_float2(0.f, 0.f);
                    if (node < N && kk < K1) {
                        if (half == 0) {
                            v = *(const float2*)(xin + (size_t)node * K1 + kk);
                        } else {
                            float s0 = 0.f, s1 = 0.f;
#pragma unroll
                            for (int j = 0; j < 16; ++j) {
                                const int off = NbLds[m * 16 + j];   // nn*K1
                                const float2 t =
                                    *(const float2*)(xin + (size_t)(off + kk));
                                s0 += t.x; s1 += t.y;
                            }
                            v = make_float2(s0, s1);
                        }
                    }
                    *(float2*)&Alds[m * APAD + 2 * kh] = v;   // aligned b64
                }
            } else {
                // scalar path (layer 0, K1 = 963): lane covers one k of row m
                const int krow  = lane;
                const int k     = k0 + krow;
                for (int m = wave; m < MB; m += 8) {
                    const int node = node0 + m;
                    float v = 0.f;
                    if (node < N && k < K1) {
                        if (half == 0) {
                            v = xin[(size_t)node * K1 + k];
                        } else {
                            float s = 0.f;
#pragma unroll
                            for (int j = 0; j < 16; ++j) {
                                const int off = NbLds[m * 16 + j];   // nn*K1
                                s += xin[(size_t)(off + k)];
                            }
                            v = s;
                        }
                    }
                    Alds[m * APAD + krow] = v;
                }
            }

            // ---- stage B chunk, k-pair interleaved: Bp[kp][2*col+(k&1)] -----
            {
                const int col4 = (tid & 63) << 2;                 // 0..252
                for (int kp = (tid >> 6); kp < KC / 2; kp += 4) { // 16 rows
                    const int k = k0 + 2 * kp;
                    float4 wa = make_float4(0.f, 0.f, 0.f, 0.f);
                    float4 wb = make_float4(0.f, 0.f, 0.f, 0.f);
                    if (k < K1)
                        wa = *(const float4*)(W + (size_t)k * DHID + col4);
                    if (k + 1 < K1)
                        wb = *(const float4*)(W + (size_t)(k + 1) * DHID + col4);
                    float* dst = &Blds[kp * BPAIR + 2 * col4];    // 16B aligned
                    dst[0] = wa.x; dst[1] = wb.x; dst[2] = wa.y; dst[3] = wb.y;
                    dst[4] = wa.z; dst[5] = wb.z; dst[6] = wa.w; dst[7] = wb.w;
                    // prefetch next chunk's weights into cache hierarchy
                    if (k0 + KC < K1)
                        __builtin_prefetch(W + (size_t)(k + KC) * DHID + col4, 0, 1);
                }
            }
            __syncthreads();

            // ---- WMMA: 8 k-steps (K=4 each) x 8 column tiles ----------------
#pragma unroll
            for (int ks = 0; ks < KC; ks += 4) {
                // A frag (16x4 f32): lanes 0-15 hold K=ks,ks+1 ; 16-31 K=ks+2,+3
                const int ka = ks + 2 * (lane >> 4);              // even
                const int ar = rt * 16 + (lane & 15);
                const v2f a = *(const v2f*)&Alds[ar * APAD + ka]; // b64
                const int kp = ka >> 1;

                // batch-load all 8 B fragments first (4x ds_load_2addr_b64),
                // then run an uninterrupted 8-WMMA burst.
                v2f b[8];
#pragma unroll
                for (int ct = 0; ct < 8; ++ct) {
                    const int colb = ch * 128 + ct * 16 + (lane & 15);
                    b[ct] = *(const v2f*)&Blds[kp * BPAIR + 2 * colb];
                }
#pragma unroll
                for (int ct = 0; ct < 8; ++ct) {
                    acc[ct] = __builtin_amdgcn_wmma_f32_16x16x4_f32(
                        /*neg_a=*/false, a, /*neg_b=*/false, b[ct],
                        /*c_mod=*/(short)0, acc[ct],
                        /*reuse_a=*/false, /*reuse_b=*/false);
                }
            }
        }
    }

    // ---- epilogue: bias + relu (+ optional residual average), store ---------
#pragma unroll
    for (int ct = 0; ct < 8; ++ct) {
        const int col = ch * 128 + ct * 16 + (lane & 15);
        const float bv = bias[col];
#pragma unroll
        for (int i = 0; i < 8; ++i) {
            // C/D layout: VGPR i -> M = i (lanes 0-15) / i+8 (lanes 16-31)
            const int row = node0 + rt * 16 + i + ((lane >> 4) << 3);
            if (row < N) {
                float v = acc[ct][i] + bv;
                v = v > 0.f ? v : 0.f;
                if (res) v = (res[(size_t)row * DHID + col] + v) * 0.5f;
                out[(size_t)row * DHID + col] = v;
            }
        }
    }
}

// ---------------------------------------------------------------------------
// coordinates = x @ W_extra(256x3) + b_extra   (memory-bound, no WMMA needed)
// ---------------------------------------------------------------------------
__global__ __launch_bounds__(256)
void coords_kernel(const float* __restrict__ x,   // [N,256]
                   const float* __restrict__ W,   // [256,3]
                   const float* __restrict__ b3,  // [3]
                   float*       __restrict__ out, // [N,3]
                   int N)
{
    const int n = blockIdx.x * blockDim.x + threadIdx.x;
    if (n >= N) return;
    float a0 = b3[0], a1 = b3[1], a2 = b3[2];
    const float* xr = x + (size_t)n * DHID;
#pragma unroll 8
    for (int k = 0; k < DHID; ++k) {
        const float xv = xr[k];                 // W[] is uniform -> SMEM loads
        a0 += xv * W[k * 3 + 0];
        a1 += xv * W[k * 3 + 1];
        a2 += xv * W[k * 3 + 2];
    }
    out[(size_t)n * 3 + 0] = a0;
    out[(size_t)n * 3 + 1] = a1;
    out[(size_t)n * 3 + 2] = a2;
}

// ---------------------------------------------------------------------------
extern "C" void kernel_launch(void* const* d_in, const int* in_sizes, int n_in,
                              void* d_out, int out_size, void* d_ws, size_t ws_size,
                              hipStream_t stream)
{
    const int*   nb       = (const int*)  d_in[0];   // [N,16]
    const float* sf       = (const float*)d_in[1];   // [N,963]
    const float* W_self0  = (const float*)d_in[2];   // [963,256]
    const float* W_neigh0 = (const float*)d_in[3];   // [963,256]
    const float* b0       = (const float*)d_in[4];   // [256]
    const float* W_self   = (const float*)d_in[5];   // [13,256,256]
    const float* W_neigh  = (const float*)d_in[6];   // [13,256,256]
    const float* bb       = (const float*)d_in[7];   // [13,256]
    const float* W_extra  = (const float*)d_in[8];   // [256,3]
    const float* b_extra  = (const float*)d_in[9];   // [3]

    const int N   = in_sizes[0] / 16;
    const int DIN = in_sizes[1] / N;                 // 963

    float* xfin   = (float*)d_out;                   // final x [N,256]
    float* coords = xfin + (size_t)N * DHID;         // [N,3]

    float* buf0 = (float*)d_ws;                      // [N,256]
    float* buf1 = buf0 + (size_t)N * DHID;           // [N,256]

    const dim3 grid((N + MB - 1) / MB), block(256);
    const size_t WSTRIDE = (size_t)DHID * DHID;      // 65536 per stacked layer

    // layer 0 : D_IN -> D_HID, relu
    gcn_layer_kernel<<<grid, block, 0, stream>>>(
        sf, nb, W_self0, W_neigh0, b0, nullptr, buf0, N, DIN);

    // 6 residual blocks of 2 GCN layers (relu each, then (temp + x)/2)
    for (int blk = 0; blk < 6; ++blk) {
        const int l1 = 2 * blk, l2 = 2 * blk + 1;
        gcn_layer_kernel<<<grid, block, 0, stream>>>(
            buf0, nb, W_self + (size_t)l1 * WSTRIDE, W_neigh + (size_t)l1 * WSTRIDE,
            bb + (size_t)l1 * DHID, nullptr, buf1, N, DHID);
        // second layer: residual = buf0, written in place (elementwise-safe)
        gcn_layer_kernel<<<grid, block, 0, stream>>>(
            buf1, nb, W_self + (size_t)l2 * WSTRIDE, W_neigh + (size_t)l2 * WSTRIDE,
            bb + (size_t)l2 * DHID, buf0, buf0, N, DHID);
    }

    // final GCN layer (stack idx 12), relu, straight into d_out
    gcn_layer_kernel<<<grid, block, 0, stream>>>(
        buf0, nb, W_self + (size_t)12 * WSTRIDE, W_neigh + (size_t)12 * WSTRIDE,
        bb + (size_t)12 * DHID, nullptr, xfin, N, DHID);

    // coordinates
    coords_kernel<<<dim3((N + 255) / 256), block, 0, stream>>>(
        xfin, W_extra, b_extra, coords, N);
}